// PGCF_49581102465511
// MI455X (gfx1250) — compile-verified
//
#include <hip/hip_runtime.h>
#include <hip/hip_bf16.h>

// LightGCN 3-layer propagation, N=250000 nodes, E=4M edges, D=64 f32.
// Feature matrix (64MB) is L2-resident on MI455X (192MB L2), so edge-parallel
// gather + atomic-f32 scatter is L2-bound, not HBM-bound.

#define TILE 512          // edges staged per block (2 per thread)
#define BLK  256
#define LPE  16           // lanes per edge (16 lanes x float4 = 64 floats)

#if defined(__HIP_DEVICE_COMPILE__)
__device__ __forceinline__ __attribute__((address_space(1))) int*
gbl_i32(const int* p) {
    return (__attribute__((address_space(1))) int*)(unsigned long long)p;
}
__device__ __forceinline__ __attribute__((address_space(3))) int*
lds_i32(int* p) {
    return (__attribute__((address_space(3))) int*)(unsigned)(unsigned long long)p;
}
#endif

// ---- degree count: deg[row[e]] += 1 (integer atomics, deterministic) ----
__global__ void k_degree(const int* __restrict__ row, unsigned* __restrict__ deg, int nE) {
    int e = blockIdx.x * blockDim.x + threadIdx.x;
    if (e < nE) atomicAdd(&deg[row[e]], 1u);
}

// ---- dinv[n] = rsqrt(deg[n] + 1)  (self-loop included; deg>0 always) ----
__global__ void k_dinv(unsigned* degi, float* dinv, int n) {
    int i = blockIdx.x * blockDim.x + threadIdx.x;
    if (i < n) {
        unsigned d = degi[i];
        dinv[i] = __frsqrt_rn((float)(d + 1u));   // in-place over the count buffer
    }
}

// ---- self-loop term, plain store (runs before edge atomics; covers all dst) ----
__global__ void k_self(const float* __restrict__ dinv,
                       const float* __restrict__ src, int sstride,
                       float* __restrict__ dst, int dstride, int n) {
    int t = blockIdx.x * blockDim.x + threadIdx.x;
    int node = t >> 4;
    int lane = t & 15;
    if (node >= n) return;
    float w = dinv[node];
    w *= w;
    const float4 v = *(const float4*)(src + (size_t)node * sstride + lane * 4);
    float4 o;
    o.x = w * v.x; o.y = w * v.y; o.z = w * v.z; o.w = w * v.w;
    *(float4*)(dst + (size_t)node * dstride + lane * 4) = o;
}

// ---- edge-parallel SpMM: dst[row] += dinv[row]*dinv[col] * src[col] ----
__global__ void __launch_bounds__(BLK) k_edges(
    const int* __restrict__ row, const int* __restrict__ col,
    const float* __restrict__ dinv,
    const float* __restrict__ src, int sstride,
    float* __restrict__ dst, int dstride, int nE)
{
    __shared__ int   s_row[TILE];
    __shared__ int   s_col[TILE];
    __shared__ float s_w[TILE];
    const int tid  = threadIdx.x;
    const int base = blockIdx.x * TILE;
    int nTile = nE - base;
    if (nTile > TILE) nTile = TILE;

    // ---- stage edge-index tile into LDS via the gfx1250 async-to-LDS path ----
#if defined(__gfx1250__) && __has_builtin(__builtin_amdgcn_global_load_async_to_lds_b32)
    {
        const int g0 = base + tid;
        const int g1 = base + BLK + tid;
        if (g0 < nE) {
            __builtin_amdgcn_global_load_async_to_lds_b32(gbl_i32(row + g0), lds_i32(&s_row[tid]), 0, 0);
            __builtin_amdgcn_global_load_async_to_lds_b32(gbl_i32(col + g0), lds_i32(&s_col[tid]), 0, 0);
        }
        if (g1 < nE) {
            __builtin_amdgcn_global_load_async_to_lds_b32(gbl_i32(row + g1), lds_i32(&s_row[BLK + tid]), 0, 0);
            __builtin_amdgcn_global_load_async_to_lds_b32(gbl_i32(col + g1), lds_i32(&s_col[BLK + tid]), 0, 0);
        }
  #if __has_builtin(__builtin_amdgcn_s_wait_asynccnt)
        __builtin_amdgcn_s_wait_asynccnt(0);
  #else
        asm volatile("s_wait_asynccnt 0" ::: "memory");
  #endif
    }
#else
    {
        const int g0 = base + tid;
        const int g1 = base + BLK + tid;
        if (g0 < nE) { s_row[tid] = row[g0]; s_col[tid] = col[g0]; }
        if (g1 < nE) { s_row[BLK + tid] = row[g1]; s_col[BLK + tid] = col[g1]; }
    }
#endif
    __syncthreads();

    // ---- per-edge weight computed once (not once per 16 lanes) ----
    for (int j = tid; j < nTile; j += BLK) {
        s_w[j] = dinv[s_row[j]] * dinv[s_col[j]];
    }
    __syncthreads();

    const int lane = tid & (LPE - 1);        // float4 slot within the 64-float row
    const int sub  = tid >> 4;               // which edge of the 16 in flight
    const float* srcl = src + lane * 4;
    float*       dstl = dst + lane * 4;

#pragma unroll 2
    for (int i = sub; i < nTile; i += 16) {
        const int   r = s_row[i];
        const int   c = s_col[i];
        const float w = s_w[i];
        // prefetch a future gather row (random 256B row, L2-resident)
        const int ip = i + 64;
        if (ip < nTile) {
            __builtin_prefetch(srcl + (size_t)s_col[ip] * sstride, 0, 0);
        }
        const float4 v = *(const float4*)(srcl + (size_t)c * sstride);
        float* p = dstl + (size_t)r * dstride;
        atomicAdd(p + 0, w * v.x);
        atomicAdd(p + 1, w * v.y);
        atomicAdd(p + 2, w * v.z);
        atomicAdd(p + 3, w * v.w);
    }
}

// ---- slot0 of all_feat = f3 (layer permutation (2,0,1)) ----
__global__ void k_copy_perm(const float* __restrict__ tail, float* __restrict__ slot0, int n) {
    int t = blockIdx.x * blockDim.x + threadIdx.x;
    int node = t >> 4;
    int lane = t & 15;
    if (node >= n) return;
    *(float4*)(slot0 + (size_t)node * 192 + lane * 4) =
        *(const float4*)(tail + (size_t)node * 64 + lane * 4);
}

extern "C" void kernel_launch(void* const* d_in, const int* in_sizes, int n_in,
                              void* d_out, int out_size, void* d_ws, size_t ws_size,
                              hipStream_t stream) {
    const int*   edge = (const int*)d_in[0];   // [2, E] int32
    const float* emb  = (const float*)d_in[1]; // [N, 64] f32
    const int E = in_sizes[0] / 2;
    const int N = in_sizes[1] / 64;
    const int* row = edge;
    const int* col = edge + E;

    unsigned* deg  = (unsigned*)d_ws;   // N counts, then overwritten by dinv
    float*    dinv = (float*)d_ws;

    float* out   = (float*)d_out;
    float* slot0 = out;                      // all_feat[:,0,:] = f3, node stride 192
    float* slot1 = out + 64;                 // all_feat[:,1,:] = f1
    float* slot2 = out + 128;                // all_feat[:,2,:] = f2
    float* tail  = out + (size_t)N * 192;    // second output: f3, contiguous stride 64

    (void)hipMemsetAsync(d_ws, 0, (size_t)N * sizeof(unsigned), stream);

    const int degBlocks  = (E + BLK - 1) / BLK;
    const int nBlocks    = (N + BLK - 1) / BLK;
    const int selfBlocks = ((N * 16) + BLK - 1) / BLK;
    const int edgeBlocks = (E + TILE - 1) / TILE;

    k_degree<<<degBlocks, BLK, 0, stream>>>(row, deg, E);
    k_dinv  <<<nBlocks,  BLK, 0, stream>>>(deg, dinv, N);

    // layer 1: emb (stride 64) -> f1 in slot1 (stride 192)
    k_self  <<<selfBlocks, BLK, 0, stream>>>(dinv, emb, 64, slot1, 192, N);
    k_edges <<<edgeBlocks, BLK, 0, stream>>>(row, col, dinv, emb, 64, slot1, 192, E);

    // layer 2: f1 (slot1, 192) -> f2 in slot2 (192)
    k_self  <<<selfBlocks, BLK, 0, stream>>>(dinv, slot1, 192, slot2, 192, N);
    k_edges <<<edgeBlocks, BLK, 0, stream>>>(row, col, dinv, slot1, 192, slot2, 192, E);

    // layer 3: f2 (slot2, 192) -> f3 in tail (64)
    k_self  <<<selfBlocks, BLK, 0, stream>>>(dinv, slot2, 192, tail, 64, N);
    k_edges <<<edgeBlocks, BLK, 0, stream>>>(row, col, dinv, slot2, 192, tail, 64, E);

    // all_feat slot0 = f3
    k_copy_perm<<<selfBlocks, BLK, 0, stream>>>(tail, slot0, N);
}